// Tem_Agg_Layer_49400713838975
// MI455X (gfx1250) — compile-verified
//
#include <hip/hip_runtime.h>
#include <hip/hip_bf16.h>

typedef __attribute__((ext_vector_type(2))) float v2f;
typedef __attribute__((ext_vector_type(8))) float v8f;

#define IN_DIM  256
#define OUT_DIM 128

// ---------------------------------------------------------------------------
// Phase 1: Z = features @ W_fc^T   via V_WMMA_F32_16X16X4_F32 (wave32)
// Each wave computes a 16(M) x 128(N) tile of Z. 4 waves / block.
// A-frag layout (ISA 7.12.2, 32-bit A 16x4): lane m = lane&15, half = lane>>4,
//   VGPR j holds A[m, k + j + 2*half]  -> one contiguous b64 load.
// B-frag mirrors with n = lane&15: B[k + j + 2*half, n] = W[n, k + j + 2*half].
// C/D layout: VGPR r, lane -> row m0 + r + 8*half, col (lane&15) + 16*t.
// ---------------------------------------------------------------------------
__global__ __launch_bounds__(128) void k_gemm(const float* __restrict__ A,
                                              const float* __restrict__ W,
                                              float* __restrict__ Z,
                                              int n_src) {
  const int lane = threadIdx.x & 31;
  const int wave = threadIdx.x >> 5;
  const int tile = blockIdx.x * 4 + wave;       // which 16-row tile
  const int m0   = tile * 16;
  if (m0 >= n_src) return;

  const int half = lane >> 4;                   // 0: lanes 0-15, 1: lanes 16-31
  const int nc   = lane & 15;                   // row (for A) / col (for B/C)

  int mrow = m0 + nc;                           // clamp A reads at the tail
  if (mrow >= n_src) mrow = n_src - 1;
  const float* __restrict__ arow = A + (size_t)mrow * IN_DIM;

  const v8f vzero = {};
  v8f acc[8];
#pragma unroll
  for (int i = 0; i < 8; ++i) acc[i] = vzero;

  for (int k = 0; k < IN_DIM; k += 4) {
    const v2f a = *(const v2f*)(arow + k + 2 * half);
#pragma unroll
    for (int tn = 0; tn < 8; ++tn) {
      const int n = tn * 16 + nc;
      const v2f b = *(const v2f*)(W + (size_t)n * IN_DIM + k + 2 * half);
      acc[tn] = __builtin_amdgcn_wmma_f32_16x16x4_f32(
          /*neg_a=*/false, a, /*neg_b=*/false, b,
          /*c_mod=*/(short)0, acc[tn], /*reuse_a=*/false, /*reuse_b=*/false);
    }
  }

#pragma unroll
  for (int r = 0; r < 8; ++r) {
    const int mm = m0 + r + 8 * half;
    if (mm < n_src) {
      float* zr = Z + (size_t)mm * OUT_DIM;
#pragma unroll
      for (int tn = 0; tn < 8; ++tn) zr[tn * 16 + nc] = acc[tn][r];
    }
  }
}

// ---------------------------------------------------------------------------
// Phase 2: score[i] = dot(Z[i,:], w_t)   (one wave per node)
// ---------------------------------------------------------------------------
__global__ __launch_bounds__(256) void k_score(const float* __restrict__ Z,
                                               const float* __restrict__ wt,
                                               float* __restrict__ score,
                                               int n_src) {
  const int w    = (blockIdx.x * blockDim.x + threadIdx.x) >> 5;
  const int lane = threadIdx.x & 31;
  if (w >= n_src) return;
  const float* zr = Z + (size_t)w * OUT_DIM;
  float s = zr[lane]      * wt[lane]
          + zr[lane + 32] * wt[lane + 32]
          + zr[lane + 64] * wt[lane + 64]
          + zr[lane + 96] * wt[lane + 96];
#pragma unroll
  for (int off = 16; off > 0; off >>= 1) s += __shfl_down(s, off, 32);
  if (lane == 0) score[w] = s;
}

// ---------------------------------------------------------------------------
// Phase 3: per-edge softmax + weighted scatter
// ---------------------------------------------------------------------------
__device__ __forceinline__ float edge_logit(const float* __restrict__ score,
                                            const float* __restrict__ t,
                                            int s, int d) {
  const float e = -fabsf(t[s] - t[d]);
  return expf(score[s] * e * 0.002f);     // exp(score * e / 500)
}

__global__ __launch_bounds__(256) void k_zero(float* __restrict__ m,
                                              float* __restrict__ denom,
                                              int n_dst) {
  const int i = blockIdx.x * blockDim.x + threadIdx.x;
  if (i < n_dst) { m[i] = 0.0f; denom[i] = 0.0f; }
}

// segment max of logits (all logits > 0  ->  int-compare == float-compare)
__global__ __launch_bounds__(256) void k_edge_max(const float* __restrict__ score,
                                                  const float* __restrict__ t,
                                                  const int* __restrict__ src,
                                                  const int* __restrict__ dst,
                                                  float* __restrict__ m, int E) {
  const int i = blockIdx.x * blockDim.x + threadIdx.x;
  if (i >= E) return;
  const float le = edge_logit(score, t, src[i], dst[i]);
  atomicMax((int*)(m + dst[i]), __float_as_int(le));
}

// segment sum of exp(logit - m[dst])
__global__ __launch_bounds__(256) void k_edge_denom(const float* __restrict__ score,
                                                    const float* __restrict__ t,
                                                    const int* __restrict__ src,
                                                    const int* __restrict__ dst,
                                                    const float* __restrict__ m,
                                                    float* __restrict__ denom, int E) {
  const int i = blockIdx.x * blockDim.x + threadIdx.x;
  if (i >= E) return;
  const int d  = dst[i];
  const float un = expf(edge_logit(score, t, src[i], d) - m[d]);
  atomicAdd(denom + d, un);
}

// out[dst] += alpha * Z[src]   (one wave per edge, lane covers 4 columns)
__global__ __launch_bounds__(256) void k_scatter(const float* __restrict__ score,
                                                 const float* __restrict__ t,
                                                 const int* __restrict__ src,
                                                 const int* __restrict__ dst,
                                                 const float* __restrict__ m,
                                                 const float* __restrict__ denom,
                                                 const float* __restrict__ Z,
                                                 float* __restrict__ out, int E) {
  const int w    = (blockIdx.x * blockDim.x + threadIdx.x) >> 5;
  const int lane = threadIdx.x & 31;
  if (w >= E) return;
  const int s = src[w];
  const int d = dst[w];
  const float le    = edge_logit(score, t, s, d);
  const float un    = expf(le - m[d]);
  const float alpha = un / fmaxf(denom[d], 1e-20f);

  const float4 zv = *(const float4*)(Z + (size_t)s * OUT_DIM + lane * 4);
  float* o = out + (size_t)d * OUT_DIM + lane * 4;
  atomicAdd(o + 0, alpha * zv.x);
  atomicAdd(o + 1, alpha * zv.y);
  atomicAdd(o + 2, alpha * zv.z);
  atomicAdd(o + 3, alpha * zv.w);
}

// ---------------------------------------------------------------------------
extern "C" void kernel_launch(void* const* d_in, const int* in_sizes, int n_in,
                              void* d_out, int out_size, void* d_ws, size_t ws_size,
                              hipStream_t stream) {
  const float* features = (const float*)d_in[0];
  const float* t        = (const float*)d_in[1];
  const float* W_fc     = (const float*)d_in[2];
  const float* w_t      = (const float*)d_in[3];
  const int*   src_idx  = (const int*)d_in[4];
  const int*   dst_idx  = (const int*)d_in[5];

  const int n_src = in_sizes[0] / IN_DIM;      // 150000
  const int E     = in_sizes[4];               // 1.6M
  const int n_dst = out_size / OUT_DIM;        // 100000

  float* zbuf  = (float*)d_ws;                       // n_src * 128
  float* score = zbuf + (size_t)n_src * OUT_DIM;     // n_src
  float* mbuf  = score + n_src;                      // n_dst
  float* denom = mbuf + n_dst;                       // n_dst
  float* out   = (float*)d_out;

  // 1) GEMM z = features @ W_fc^T  (WMMA f32 16x16x4)
  const int ntile = (n_src + 15) / 16;
  k_gemm<<<(ntile + 3) / 4, 128, 0, stream>>>(features, W_fc, zbuf, n_src);

  // 2) per-node temporal score
  k_score<<<(n_src + 7) / 8, 256, 0, stream>>>(zbuf, w_t, score, n_src);

  // 3) residual init (out = z[:n_dst]) + zero segment accumulators
  hipMemcpyAsync(out, zbuf, (size_t)n_dst * OUT_DIM * sizeof(float),
                 hipMemcpyDeviceToDevice, stream);
  k_zero<<<(n_dst + 255) / 256, 256, 0, stream>>>(mbuf, denom, n_dst);

  // 4) segment max of logits
  k_edge_max<<<(E + 255) / 256, 256, 0, stream>>>(score, t, src_idx, dst_idx, mbuf, E);

  // 5) segment sum of shifted exps
  k_edge_denom<<<(E + 255) / 256, 256, 0, stream>>>(score, t, src_idx, dst_idx,
                                                    mbuf, denom, E);

  // 6) weighted scatter: out[dst] += alpha * z[src]  (one wave per edge)
  k_scatter<<<(E + 7) / 8, 256, 0, stream>>>(score, t, src_idx, dst_idx,
                                             mbuf, denom, zbuf, out, E);
}